// AttrPredLoss_40_40261023433195
// MI455X (gfx1250) — compile-verified
//
#include <hip/hip_runtime.h>

typedef float  v2f __attribute__((ext_vector_type(2)));
typedef float  v4f __attribute__((ext_vector_type(4)));
typedef float  v8f __attribute__((ext_vector_type(8)));
typedef int    v2i __attribute__((ext_vector_type(2)));

#define NBLOCKS  1024
#define NTHREADS 256

// Sum of all 32 lane values of v, exact f32 FMA precision, via
// V_WMMA_F32_16X16X4_F32 with A = ones(16x4), B slots = per-lane partials.
// D[m][n] = sum_k B[k][n] for every m, so summing D row 0 over its 16 lanes
// (shuffle-xor 1,2,4,8 stays within lanes 0..15) yields the total of all 64
// B slots == sum of the 32 lane inputs. Result valid in lane 0.
__device__ __forceinline__ float wave32_sum_wmma(float v) {
  v2f a; a.x = 1.0f; a.y = 1.0f;        // A-matrix: all ones
  v2f b; b.x = v;    b.y = 0.0f;        // B-matrix: lane partial + zero slot
  v8f c = {};
#if __has_builtin(__builtin_amdgcn_wmma_f32_16x16x4_f32)
  c = __builtin_amdgcn_wmma_f32_16x16x4_f32(
      /*neg_a=*/false, a, /*neg_b=*/false, b,
      /*c_mod=*/(short)0, c, /*reuse_a=*/false, /*reuse_b=*/false);
#else
  asm volatile("v_wmma_f32_16x16x4_f32 %0, %1, %2, 0"
               : "+v"(c) : "v"(a), "v"(b));
#endif
  float col = c[0];                      // lanes 0..15: colsum[N=lane]
  col += __shfl_xor(col, 1, 32);
  col += __shfl_xor(col, 2, 32);
  col += __shfl_xor(col, 4, 32);
  col += __shfl_xor(col, 8, 32);
  return col;                            // lane 0: total
}

__global__ __launch_bounds__(NTHREADS)
void focal_partial_kernel(const float* __restrict__ pred,
                          const int*   __restrict__ label,
                          float* __restrict__ partial, int B) {
  // SEL - 4 (we load columns 4..39 of each 40-wide row, 16B aligned)
  constexpr int SELm4[18] = {0,4,5,7,9,10,11,12,13,14,16,18,19,22,25,26,27,32};

  float acc = 0.0f;
  const int stride = gridDim.x * NTHREADS;
  for (int row = blockIdx.x * NTHREADS + threadIdx.x; row < B; row += stride) {
    // pred row base = row*160 bytes; +16B => cols 4..39 as nine aligned b128 loads
    const v4f* pr = (const v4f*)(pred + (size_t)row * 40 + 4);
    float r[36];
#pragma unroll
    for (int i = 0; i < 9; ++i) {
      v4f v = __builtin_nontemporal_load(pr + i);
      r[4*i+0] = v.x; r[4*i+1] = v.y; r[4*i+2] = v.z; r[4*i+3] = v.w;
    }
    // label row base = row*72 bytes (8B aligned) => nine b64 loads
    const v2i* lr = (const v2i*)(label + (size_t)row * 18);
    int l[18];
#pragma unroll
    for (int i = 0; i < 9; ++i) {
      v2i v = __builtin_nontemporal_load(lr + i);
      l[2*i+0] = v.x; l[2*i+1] = v.y;
    }

    float bce = 0.0f;   // sum of selected log-prob terms (pre-negation)
    float fw  = 0.0f;   // sum of focal weights
#pragma unroll
    for (int j = 0; j < 18; ++j) {
      float p   = r[SELm4[j]];
      bool  pos = (l[j] == 1);
      float lg  = pos ? __logf(p) : __logf(1.0f - p);   // log / log1p(-p)
      lg = fmaxf(lg, -100.0f);
      bce += lg;
      float w     = pos ? (1.0f - p) : p;
      float alpha = pos ? 0.8f : 0.2f;
      fw = fmaf(alpha * w, w, fw);
    }
    // contrib = fw_sum * (-bce_sum / 18)
    acc = fmaf(fw, bce * (-1.0f / 18.0f), acc);
  }

  // wave32 shuffle reduction, then 8 wave-leaders through LDS
  acc += __shfl_xor(acc, 16, 32);
  acc += __shfl_xor(acc,  8, 32);
  acc += __shfl_xor(acc,  4, 32);
  acc += __shfl_xor(acc,  2, 32);
  acc += __shfl_xor(acc,  1, 32);

  __shared__ float swave[NTHREADS / 32];
  const int lane = threadIdx.x & 31;
  const int wid  = threadIdx.x >> 5;
  if (lane == 0) swave[wid] = acc;
  __syncthreads();
  if (threadIdx.x == 0) {
    float t = 0.0f;
#pragma unroll
    for (int i = 0; i < NTHREADS / 32; ++i) t += swave[i];
    partial[blockIdx.x] = t;
  }
}

__global__ __launch_bounds__(NTHREADS)
void focal_final_kernel(const float* __restrict__ partial,
                        float* __restrict__ out) {
  __shared__ float sdata[NTHREADS];
  float s = 0.0f;
#pragma unroll
  for (int k = 0; k < NBLOCKS / NTHREADS; ++k)
    s += partial[threadIdx.x + k * NTHREADS];
  sdata[threadIdx.x] = s;
  __syncthreads();
  if (threadIdx.x < 128) sdata[threadIdx.x] += sdata[threadIdx.x + 128];
  __syncthreads();
  if (threadIdx.x < 64)  sdata[threadIdx.x] += sdata[threadIdx.x + 64];
  __syncthreads();

  // All waves execute the WMMA with full EXEC (ISA requirement); only wave 0
  // carries real data (other waves feed zeros and discard the result).
  float v = (threadIdx.x < 32)
              ? (sdata[threadIdx.x] + sdata[threadIdx.x + 32])
              : 0.0f;
  float tot = wave32_sum_wmma(v);
  if (threadIdx.x == 0) out[0] = tot;
}

extern "C" void kernel_launch(void* const* d_in, const int* in_sizes, int n_in,
                              void* d_out, int out_size, void* d_ws, size_t ws_size,
                              hipStream_t stream) {
  const float* pred  = (const float*)d_in[0];
  const int*   label = (const int*)d_in[1];
  float* out     = (float*)d_out;
  float* partial = (float*)d_ws;            // NBLOCKS floats = 4 KB scratch
  const int B = in_sizes[1] / 18;           // rows

  focal_partial_kernel<<<NBLOCKS, NTHREADS, 0, stream>>>(pred, label, partial, B);
  focal_final_kernel<<<1, NTHREADS, 0, stream>>>(partial, out);
}